// PotentialModel_37512244363969
// MI455X (gfx1250) — compile-verified
//
#include <hip/hip_runtime.h>
#include <math.h>

typedef __attribute__((ext_vector_type(2))) float v2f;
typedef __attribute__((ext_vector_type(8))) float v8f;

#define N_SRC 8192
#define M_PTS 16384
#define N_TILES (N_SRC / 16)            // 512 source tiles
#define N_CHUNKS 8
#define CHUNK (N_SRC / N_CHUNKS)        // 1024 sources per wave
#define TILES_PER_CHUNK (CHUNK / 16)    // 64 WMMA tiles per wave
#define WAVES_PER_BLOCK 8               // 256 threads = 8 wave32

// Init: zero the output (harness poisons it with 0xAA) and pre-pack the WMMA
// A-operand in its exact lane layout:
//   Apk[tile][lane] = (v2f) A-matrix registers for V_WMMA_F32_16X16X4_F32
//   lanes 0-15  (row = lane):    (-2*rx, -2*ry)      -> K0, K1
//   lanes 16-31 (row = lane-16): (-2*rz,  |r|^2)     -> K2, K3
// so the hot loop needs only one b64 load per tile for the whole A matrix.
__global__ __launch_bounds__(256) void pot_init_kernel(const float* __restrict__ r_p,
                                                       v2f* __restrict__ Apk,
                                                       float* __restrict__ out) {
    int i = blockIdx.x * blockDim.x + threadIdx.x;   // 16384 threads
    if (i < N_TILES * 32) {
        const int tile = i >> 5;
        const int lane = i & 31;
        const int half = lane >> 4;
        const int idx  = lane & 15;
        const int n    = tile * 16 + idx;
        const float x = r_p[3 * n + 0];
        const float y = r_p[3 * n + 1];
        const float z = r_p[3 * n + 2];
        v2f a;
        if (half == 0) { a.x = -2.0f * x; a.y = -2.0f * y; }
        else           { a.x = -2.0f * z; a.y = x * x + y * y + z * z; }
        Apk[i] = a;
    }
    if (i < M_PTS) out[i] = 0.0f;
}

// One wave per (m_tile of 16 queries, n_chunk of 1024 sources).
// sq(n,m) = r2[n] + q2[m] - 2 r.q is produced wholesale by
// V_WMMA_F32_16X16X4_F32 with:
//   A[i][:] = (-2rx, -2ry, -2rz, r2[n_i])   (K padded to 4, pre-packed)
//   B[:][j] = ( qx,   qy,   qz,  1      )
//   C[i][j] =   q2[m_j]
// then potential contribution = b[n_i] * rsqrt(sq), reduced over rows.
__global__ __launch_bounds__(256) void pot_wmma_kernel(const float* __restrict__ b_n,
                                                       const float* __restrict__ coord,
                                                       const v2f* __restrict__ Apk,
                                                       float* __restrict__ out) {
    const int lane      = threadIdx.x & 31;
    const int waveInBlk = threadIdx.x >> 5;
    const int w         = blockIdx.x * WAVES_PER_BLOCK + waveInBlk;
    const int mTile     = w / N_CHUNKS;   // 0..1023
    const int chunk     = w % N_CHUNKS;   // 0..7
    const int half      = lane >> 4;      // row-half selector
    const int idx       = lane & 15;      // A row / B,C,D column index

    // ---- loop-invariant B and C operands for this wave's 16 query points ----
    const int   m  = mTile * 16 + idx;
    const float CZ = 0.3989422804014327f;             // 1/sqrt(2*pi) depth offset
    const float qx = coord[3 * m + 0];
    const float qy = coord[3 * m + 1];
    const float qz = coord[3 * m + 2] + CZ;
    const float q2 = qx * qx + qy * qy + qz * qz;

    v2f bmat;                                          // B 4x16: lane = column
    bmat.x = half ? qz : qx;                           // K2 : K0
    bmat.y = half ? 1.0f : qy;                         // K3 : K1
    v8f cmat;
#pragma unroll
    for (int k = 0; k < 8; ++k) cmat[k] = q2;          // C[i][j] = q2[m_j]

    const int n0base   = chunk * CHUNK;
    const v2f* ap      = Apk + (n0base / 16) * 32 + lane;   // stride 32 per tile
    const float* bbase = b_n + n0base + half * 8;
    const float  INF   = __builtin_inff();

    float acc0 = 0.0f, acc1 = 0.0f;

    for (int t = 0; t < TILES_PER_CHUNK; ++t) {
        const v2f amat = ap[t * 32];                   // one b64: whole A matrix

        // D[i][j] = sq(n_i, m_j) -- one matrix op per 256 pairs
        const v8f d = __builtin_amdgcn_wmma_f32_16x16x4_f32(
            false, amat, false, bmat, (short)0, cmat, false, false);

        // b weights for this lane's 8 rows (rows half*8 .. half*8+7), 16B aligned
        const float4* bp  = (const float4*)(bbase + t * 16);
        const float4  bv0 = bp[0];
        const float4  bv1 = bp[1];
        const float   bw[8] = {bv0.x, bv0.y, bv0.z, bv0.w,
                               bv1.x, bv1.y, bv1.z, bv1.w};

#pragma unroll
        for (int k = 0; k < 8; ++k) {
            // clamp to [0, +inf) in a single v_med3_f32 (reference: maximum(sq,0))
            const float sq = __builtin_amdgcn_fmed3f(d[k], 0.0f, INF);
            const float rs = __builtin_amdgcn_rsqf(sq);   // v_rsq_f32
            if (k & 1) acc1 = fmaf(bw[k], rs, acc1);
            else       acc0 = fmaf(bw[k], rs, acc0);
        }
    }

    float acc = acc0 + acc1;
    // lanes j and j+16 hold the row-half partial sums of column j
    acc += __shfl_xor(acc, 16, 32);

    const float INV2PI = 0.15915494309189535f;
    if (lane < 16) {
        atomicAdd(&out[mTile * 16 + idx], acc * INV2PI);
    }
}

extern "C" void kernel_launch(void* const* d_in, const int* in_sizes, int n_in,
                              void* d_out, int out_size, void* d_ws, size_t ws_size,
                              hipStream_t stream) {
    const float* b_n   = (const float*)d_in[0];   // (8192,)
    const float* r_p   = (const float*)d_in[1];   // (8192, 3)
    const float* coord = (const float*)d_in[2];   // (16384, 3)
    float*       out   = (float*)d_out;           // (16384,)
    v2f*         Apk   = (v2f*)d_ws;              // 512 tiles * 32 lanes * 8B = 128KB

    (void)in_sizes; (void)n_in; (void)out_size; (void)ws_size;

    pot_init_kernel<<<(M_PTS + 255) / 256, 256, 0, stream>>>(r_p, Apk, out);

    const int totalWaves = (M_PTS / 16) * N_CHUNKS;           // 8192 waves
    const int blocks     = totalWaves / WAVES_PER_BLOCK;      // 1024 blocks
    pot_wmma_kernel<<<blocks, 256, 0, stream>>>(b_n, coord, Apk, out);
}